// Net_19224273617064
// MI455X (gfx1250) — compile-verified
//
#include <hip/hip_runtime.h>
#include <math.h>

// Problem dimensions (fixed by the reference)
#define Bsz     2
#define Nn      400
#define Ff      246
#define Cc      32
#define LBL     1440
#define NODE_IN 310          // F + 2*C
#define ROWS    (Bsz * Nn)   // 800 flattened (b, n) rows

typedef float v2f __attribute__((ext_vector_type(2)));
typedef float v8f __attribute__((ext_vector_type(8)));

// ---------------------------------------------------------------------------
// fp32 GEMM on the CDNA5 matrix pipe: C[M,N] = A[M,K] @ B[K,N] (+bias).
// One wave computes a 16x32 D tile (two 16x16 WMMA accumulators sharing the
// A operand). Hot loop is branch-free: out-of-range rows/cols are CLAMPED
// (they only feed D elements we never store), and the K%4 tail is peeled
// into a single guarded iteration. A's two k-values per lane are contiguous,
// so A is fetched as one b64 load per step.
// ---------------------------------------------------------------------------
__global__ __launch_bounds__(32)
void gemm_wmma_f32(const float* __restrict__ A, int lda,
                   const float* __restrict__ Bm, int ldb,
                   const float* __restrict__ bias,
                   float* __restrict__ Cm, int ldc,
                   int M, int N, int K) {
    const int lane = threadIdx.x;        // 0..31 (wave32)
    const int half = lane >> 4;          // K split for A/B operands
    const int l15  = lane & 15;
    const int tm   = blockIdx.y;         // 16-high tile
    const int tn   = blockIdx.x;         // 32-wide tile

    const int row  = tm * 16 + l15;
    const int amr  = min(row, M - 1);        // clamped A row (safe: unstored)
    const int col0 = tn * 32 + l15;
    const int col1 = col0 + 16;
    const int bc0  = min(col0, N - 1);       // clamped B cols (safe: unstored)
    const int bc1  = min(col1, N - 1);

    const float* __restrict__ Arow = A  + (size_t)amr * lda + half * 2;
    const float* __restrict__ Bp   = Bm + (size_t)(half * 2) * ldb;

    v8f acc0 = {0.f,0.f,0.f,0.f,0.f,0.f,0.f,0.f};
    v8f acc1 = {0.f,0.f,0.f,0.f,0.f,0.f,0.f,0.f};

    const int Ka = K & ~3;
    int kk = 0;
#pragma unroll 2
    for (; kk < Ka; kk += 4) {
        const v2f a = *(const v2f*)(Arow + kk);          // A[amr][kk+2h .. +1]
        v2f b0, b1;
        const int kb = kk * ldb;
        b0.x = Bp[kb + bc0];       b0.y = Bp[kb + ldb + bc0];
        b1.x = Bp[kb + bc1];       b1.y = Bp[kb + ldb + bc1];
        acc0 = __builtin_amdgcn_wmma_f32_16x16x4_f32(false, a, false, b0,
                                                     (short)0, acc0, false, false);
        acc1 = __builtin_amdgcn_wmma_f32_16x16x4_f32(false, a, false, b1,
                                                     (short)0, acc1, false, false);
    }
    if (kk < K) {                         // peeled tail: K % 4 in {1,2,3}
        const int k0 = kk + half * 2;
        const int k1 = k0 + 1;
        v2f a, b0, b1;
        a.x  = (k0 < K) ? A[(size_t)amr * lda + k0]  : 0.f;
        a.y  = (k1 < K) ? A[(size_t)amr * lda + k1]  : 0.f;
        b0.x = (k0 < K) ? Bm[(size_t)k0 * ldb + bc0] : 0.f;
        b0.y = (k1 < K) ? Bm[(size_t)k1 * ldb + bc0] : 0.f;
        b1.x = (k0 < K) ? Bm[(size_t)k0 * ldb + bc1] : 0.f;
        b1.y = (k1 < K) ? Bm[(size_t)k1 * ldb + bc1] : 0.f;
        acc0 = __builtin_amdgcn_wmma_f32_16x16x4_f32(false, a, false, b0,
                                                     (short)0, acc0, false, false);
        acc1 = __builtin_amdgcn_wmma_f32_16x16x4_f32(false, a, false, b1,
                                                     (short)0, acc1, false, false);
    }

    // D layout: VGPR r -> row = tm*16 + half*8 + r, col = tn*32 + (lane&15) [+16]
    if (col0 < N) {
        const float bb = bias ? bias[col0] : 0.f;
#pragma unroll
        for (int r = 0; r < 8; ++r) {
            const int rr = tm * 16 + half * 8 + r;
            if (rr < M) Cm[(size_t)rr * ldc + col0] = acc0[r] + bb;
        }
    }
    if (col1 < N) {
        const float bb = bias ? bias[col1] : 0.f;
#pragma unroll
        for (int r = 0; r < 8; ++r) {
            const int rr = tm * 16 + half * 8 + r;
            if (rr < M) Cm[(size_t)rr * ldc + col1] = acc1[r] + bb;
        }
    }
}

// ---------------------------------------------------------------------------
// Pair aggregation. s[b,i,j,c] = P[b,i,c] + Q[b,j,c] + e_ij*w3 + e_ji*w4 + bs
// -> PReLU -> mask by a -> sigmoid gate (cross-lane dot over 32 channels)
// -> accumulate into xcat[row, 246+c] (dir=0, sum over j) or
//    xcat[row, 278+c] (dir=1, sum over i).  One wave per (b, node); lane = c.
// The `a != 0` branch is wave-uniform (all lanes share (b,i,j)).
// ---------------------------------------------------------------------------
__device__ __forceinline__ float wave_sum32(float v) {
#pragma unroll
    for (int off = 16; off > 0; off >>= 1) v += __shfl_xor(v, off, 32);
    return v;
}

__global__ __launch_bounds__(32)
void pair_agg(const float* __restrict__ x,  const float* __restrict__ a,
              const float* __restrict__ e,
              const float* __restrict__ Wstack, const float* __restrict__ bstack,
              const float* __restrict__ alpha,
              const float* __restrict__ Watt,   const float* __restrict__ batt,
              const float* __restrict__ P, const float* __restrict__ Q,
              float* __restrict__ xcat, int dir) {
    const int r    = blockIdx.x;          // flattened (b, n0)
    const int b    = r / Nn;
    const int n0   = r % Nn;
    const int lane = threadIdx.x;         // channel c

    const float w3 = Wstack[492 * Cc + lane];   // e_ij coefficient
    const float w4 = Wstack[493 * Cc + lane];   // e_ji coefficient
    const float bs = bstack[lane];
    const float al = alpha[lane];
    const float wa = Watt[lane];
    const float ba = batt[0];
    const float fixedPQ = (dir == 0) ? P[r * Cc + lane] : Q[r * Cc + lane];

    float acc = 0.f;
    for (int m = 0; m < Nn; ++m) {
        const int i  = (dir == 0) ? n0 : m;
        const int j  = (dir == 0) ? m  : n0;
        const int ij = (b * Nn + i) * Nn + j;   // S == 1: e index == a index
        const int ji = (b * Nn + j) * Nn + i;
        const float av = a[ij];                 // uniform across the wave
        if (av != 0.f) {
            float s = fixedPQ
                    + ((dir == 0) ? Q[(b * Nn + j) * Cc + lane]
                                  : P[(b * Nn + i) * Cc + lane])
                    + e[ij] * w3 + e[ji] * w4 + bs;
            s = (s >= 0.f) ? s : al * s;        // PReLU
            s *= av;                            // edge mask (a is 0/1 float)
            const float t = wave_sum32(s * wa) + ba;
            const float g = 1.f / (1.f + __expf(-t));
            acc += s * g;
        }
    }
    xcat[r * NODE_IN + ((dir == 0) ? Ff : (Ff + Cc)) + lane] = acc;

    if (dir == 0) {   // also materialize the x part of concat([x, agg_i, agg_j])
        for (int f = lane; f < Ff; f += 32)
            xcat[r * NODE_IN + f] = x[r * Ff + f];
    }
}

// ---------------------------------------------------------------------------
extern "C" void kernel_launch(void* const* d_in, const int* in_sizes, int n_in,
                              void* d_out, int out_size, void* d_ws, size_t ws_size,
                              hipStream_t stream) {
    const float* x       = (const float*)d_in[0];
    const float* a       = (const float*)d_in[1];
    const float* e       = (const float*)d_in[2];
    const float* W_stack = (const float*)d_in[3];
    const float* b_stack = (const float*)d_in[4];
    const float* alpha   = (const float*)d_in[5];
    const float* W_att_i = (const float*)d_in[6];
    const float* b_att_i = (const float*)d_in[7];
    const float* W_att_j = (const float*)d_in[8];
    const float* b_att_j = (const float*)d_in[9];
    const float* W_node  = (const float*)d_in[10];
    const float* b_node  = (const float*)d_in[11];
    const float* W_out   = (const float*)d_in[12];
    const float* b_out   = (const float*)d_in[13];
    float*       out     = (float*)d_out;

    // Workspace (floats): P(800x32) | Q(800x32) | xcat(800x310) | xnew(800x246)
    float* P    = (float*)d_ws;
    float* Q    = P + ROWS * Cc;
    float* xcat = Q + ROWS * Cc;
    float* xnew = xcat + ROWS * NODE_IN;

    const dim3 blk(32);

    // P = x @ W_stack[0:246]      (M=800, N=32 -> one 32-wide tile, K=246)
    gemm_wmma_f32<<<dim3(1, ROWS / 16), blk, 0, stream>>>(
        x, Ff, W_stack, Cc, nullptr, P, Cc, ROWS, Cc, Ff);
    // Q = x @ W_stack[246:492]
    gemm_wmma_f32<<<dim3(1, ROWS / 16), blk, 0, stream>>>(
        x, Ff, W_stack + Ff * Cc, Cc, nullptr, Q, Cc, ROWS, Cc, Ff);

    // agg_in  (sum over j, gate W_att_i) -> xcat[:,246:278]; also copies x rows
    pair_agg<<<ROWS, blk, 0, stream>>>(x, a, e, W_stack, b_stack, alpha,
                                       W_att_i, b_att_i, P, Q, xcat, 0);
    // agg_out (sum over i, gate W_att_j) -> xcat[:,278:310]
    pair_agg<<<ROWS, blk, 0, stream>>>(x, a, e, W_stack, b_stack, alpha,
                                       W_att_j, b_att_j, P, Q, xcat, 1);

    // x_new = xcat @ W_node + b_node     (800x310 @ 310x246; N tiles = 8)
    gemm_wmma_f32<<<dim3((Ff + 31) / 32, ROWS / 16), blk, 0, stream>>>(
        xcat, NODE_IN, W_node, Ff, b_node, xnew, Ff, ROWS, Ff, NODE_IN);

    // out = x_new @ W_out + b_out        (800x246 @ 246x1440; N tiles = 45)
    gemm_wmma_f32<<<dim3(LBL / 32, ROWS / 16), blk, 0, stream>>>(
        xnew, Ff, W_out, LBL, b_out, out, LBL, ROWS, LBL, Ff);
}